// DWT_2D_Low_19164144075604
// MI455X (gfx1250) — compile-verified
//
#include <hip/hip_runtime.h>

typedef __attribute__((ext_vector_type(2))) float v2f;
typedef __attribute__((ext_vector_type(8))) float v8f;

#define IMG_H 512
#define IMG_W 512
#define OUT_P 256
#define OUT_Q 256
#define LDS_STRIDE 52   // 16x48 T tile padded: 8*52 % 64 == 32 -> no bank conflict between half-waves

// band coefficient: matrix_h[i][kk_local] = f[kk - 2*i] if 0 <= kk-2i < 4 else 0
__device__ __forceinline__ float band_coef(const float f[4], int i, int kk) {
  int d = kk - 2 * i;
  return (d >= 0 && d < 4) ? f[d] : 0.0f;
}

__device__ __forceinline__ int clampi(int v, int lo, int hi) {
  int t = v < lo ? lo : v;
  return t > hi ? hi : t;
}

__global__ __launch_bounds__(256) void dwt2d_low_wmma(
    const float* __restrict__ x,   // (512 images, 512, 512)
    const float* __restrict__ mh,  // (256, 512) banded low filter
    float* __restrict__ out)       // (512 images, 256, 256)
{
  // per-wave T staging: C-layout -> A-layout transpose through LDS
  __shared__ float Tlds[8][16 * LDS_STRIDE];

  const int lane = threadIdx.x & 31;
  const int wave = threadIdx.x >> 5;
  // tile id is wave-uniform: pin it to an SGPR so tile math + the
  // interior/boundary branch are pure SALU (EXEC untouched around WMMA).
  const int tileId = __builtin_amdgcn_readfirstlane((int)blockIdx.x * 8 + wave);
  const int img = tileId >> 8;                // 512 images
  const int tp  = (tileId >> 4) & 15;         // 16 row tiles
  const int tq  = tileId & 15;                // 16 col tiles

  const float* __restrict__ Ximg = x + (size_t)img * IMG_H * IMG_W;
  float* __restrict__ Oimg = out + (size_t)img * OUT_P * OUT_Q;

  // filter taps live densely in matrix_h row 1, cols 1..4 (h = 2*1-1+k)
  float f[4];
  f[0] = mh[IMG_W + 1];
  f[1] = mh[IMG_W + 2];
  f[2] = mh[IMG_W + 3];
  f[3] = mh[IMG_W + 4];

  const int nloc = lane & 15;            // M index (A) / N index (B)
  const int koff = (lane >> 4) << 1;     // lanes 16-31 carry K+2,K+3

  // Per-lane band constants for K-chunk c (K = 4c .. 4c+3).
  // Serve simultaneously as the stage-1 A operand (row filter) and the
  // stage-2 B operand (column filter) thanks to mirrored A/B layouts.
  v2f cf[9];
  #pragma unroll
  for (int c = 0; c < 9; ++c) {
    cf[c].x = band_coef(f, nloc, 4 * c + koff);
    cf[c].y = band_coef(f, nloc, 4 * c + koff + 1);
  }

  const int hbase = 32 * tp - 1;  // first input row touched by this tile
  const int wbase = 32 * tq - 1;  // first input col touched by this tile

  // ---------- stage 1: T(16 x 48) = A_band(16 x 36) * X(36 x 48) ----------
  v8f accT[3] = {{}, {}, {}};

  if (tp >= 1 && tp <= 14 && tq >= 1 && tq <= 14) {
    // Interior tile: rows hbase..hbase+35 and cols wbase..wbase+47 are all
    // in-range. One per-lane base address; every load gets a compile-time
    // immediate offset -> clause-able unconditional global_load_b32s.
    const float* __restrict__ px =
        Ximg + (size_t)(hbase + koff) * IMG_W + (wbase + nloc);
    #pragma unroll
    for (int c = 0; c < 9; ++c) {
      #pragma unroll
      for (int jt = 0; jt < 3; ++jt) {
        v2f b;
        b.x = px[c * 4 * IMG_W + jt * 16];
        b.y = px[c * 4 * IMG_W + jt * 16 + IMG_W];
        accT[jt] = __builtin_amdgcn_wmma_f32_16x16x4_f32(
            false, cf[c], false, b, (short)0, accT[jt], false, false);
      }
    }
  } else {
    // Boundary tile: clamp coordinates (always-valid address), load
    // unconditionally, then select 0 for out-of-range taps. Straight-line,
    // no exec-mask diamonds.
    #pragma unroll
    for (int c = 0; c < 9; ++c) {
      const int r0 = hbase + 4 * c + koff;
      const int r1 = r0 + 1;
      const int rc0 = clampi(r0, 0, IMG_H - 1);
      const int rc1 = clampi(r1, 0, IMG_H - 1);
      const bool r0ok = (r0 == rc0);
      const bool r1ok = (r1 == rc1);
      #pragma unroll
      for (int jt = 0; jt < 3; ++jt) {
        const int col = wbase + 16 * jt + nloc;
        const int colc = clampi(col, 0, IMG_W - 1);
        const bool cok = (col == colc);
        const float v0 = Ximg[(size_t)rc0 * IMG_W + colc];
        const float v1 = Ximg[(size_t)rc1 * IMG_W + colc];
        v2f b;
        b.x = (r0ok && cok) ? v0 : 0.0f;
        b.y = (r1ok && cok) ? v1 : 0.0f;
        accT[jt] = __builtin_amdgcn_wmma_f32_16x16x4_f32(
            false, cf[c], false, b, (short)0, accT[jt], false, false);
      }
    }
  }

  // ---------- spill T (C-layout) to LDS ----------
  {
    const int rowoff = (lane >> 4) << 3;  // C-layout: lanes 16-31 hold rows M+8
    #pragma unroll
    for (int jt = 0; jt < 3; ++jt) {
      #pragma unroll
      for (int r = 0; r < 8; ++r) {
        Tlds[wave][(r + rowoff) * LDS_STRIDE + 16 * jt + nloc] = accT[jt][r];
      }
    }
  }
  __syncthreads();

  // ---------- stage 2: Out(16 x 16) = T(16 x 36) * B_band(36 x 16) ----------
  v8f accO = {};
  #pragma unroll
  for (int c = 0; c < 9; ++c) {
    // A-layout: lane nloc, K = 4c+koff, 4c+koff+1 (8B-aligned -> ds_load_b64)
    const v2f a = *(const v2f*)&Tlds[wave][nloc * LDS_STRIDE + 4 * c + koff];
    accO = __builtin_amdgcn_wmma_f32_16x16x4_f32(
        false, a, false, cf[c], (short)0, accO, false, false);
  }

  // ---------- store 16x16 output tile from C-layout ----------
  const int p0 = 16 * tp;
  const int q0 = 16 * tq;
  const int rowoff = (lane >> 4) << 3;
  float* __restrict__ po = Oimg + (size_t)(p0 + rowoff) * OUT_Q + q0 + nloc;
  #pragma unroll
  for (int r = 0; r < 8; ++r) {
    po[r * OUT_Q] = accO[r];
  }
}

extern "C" void kernel_launch(void* const* d_in, const int* in_sizes, int n_in,
                              void* d_out, int out_size, void* d_ws, size_t ws_size,
                              hipStream_t stream) {
  (void)in_sizes; (void)n_in; (void)d_ws; (void)ws_size; (void)out_size;
  const float* x  = (const float*)d_in[0];   // (8,64,512,512) fp32
  const float* mh = (const float*)d_in[1];   // (256,512) fp32
  // d_in[2] (matrix_h_t) is the transpose of the same band filter; unused.
  float* out = (float*)d_out;                // (8,64,256,256) fp32

  // 512 images * 256 tiles = 131072 waves, 8 waves (256 threads) per block
  dim3 grid(131072 / 8);
  dim3 block(256);
  hipLaunchKernelGGL(dwt2d_low_wmma, grid, block, 0, stream, x, mh, out);
}